// GeneratorMixture_86835648790546
// MI455X (gfx1250) — compile-verified
//
#include <hip/hip_runtime.h>
#include <hip/hip_bf16.h>

// Problem constants (match reference).
#define BB   256   // batch
#define KK   2     // top_k
#define EE   16    // experts
#define DD   512   // dim
#define NPAIR (BB*KK)          // 512 (b,k) pairs
#define IVLEN (NPAIR*DD)       // 262144 floats per intermediate array

typedef float v2f __attribute__((ext_vector_type(2)));
typedef float v8f __attribute__((ext_vector_type(8)));

// ---------------------------------------------------------------------------
// Kernel A: deterministic expert binning. Thread e (0..15) scans all 512
// pairs for weight_indices==e; threads 16..31 do the same for bias_indices.
// Sequential scan => bucket order is deterministic (no atomics).
// ---------------------------------------------------------------------------
__global__ void gen_bin(const int* __restrict__ widx, const int* __restrict__ bidx,
                        int* __restrict__ wcnt, int* __restrict__ bcnt,
                        int* __restrict__ wbkt, int* __restrict__ bbkt) {
    int t = threadIdx.x;
    if (t < EE) {
        int e = t, c = 0;
        for (int p = 0; p < NPAIR; ++p)
            if (widx[p] == e) wbkt[e * NPAIR + c++] = p;
        wcnt[e] = c;
    } else if (t < 2 * EE) {
        int e = t - EE, c = 0;
        for (int p = 0; p < NPAIR; ++p)
            if (bidx[p] == e) bbkt[e * NPAIR + c++] = p;
        bcnt[e] = c;
    }
}

// ---------------------------------------------------------------------------
// Kernel B: per-expert GEMMs using V_WMMA_F32_16X16X4_F32 (wave32).
// grid = (32 N-tiles, 16 experts, 4 banks), block = 32 (one wave per block,
// EXEC all-1s around the WMMA as the ISA requires).
// bank 0/1/2 -> input/output/diagonal weight banks with weight buckets,
// bank 3     -> bias bank with bias buckets.
// A-matrix (16x4 f32) per ISA layout: lanes 0-15 hold K=0(K=1 in v1),
// lanes 16-31 hold K=2/K=3 for the same M rows. B-matrix (4x16) mirrored.
// C/D 16x16 f32: vgpr r -> M=r (lanes 0-15) / M=8+r (lanes 16-31), N=lane&15.
//
// Rows m >= mrem of the A tile are fed with a clamped (valid) bucket row
// instead of zeros: D-row m depends only on A-row m, and those rows are
// never stored, so no masking is needed in the inner loop.
// ---------------------------------------------------------------------------
__global__ __launch_bounds__(32)
void gen_expert_gemm(const float* __restrict__ x,
                     const float* __restrict__ w_in,
                     const float* __restrict__ w_out,
                     const float* __restrict__ w_diag,
                     const float* __restrict__ w_bias,
                     const int* __restrict__ wcnt, const int* __restrict__ bcnt,
                     const int* __restrict__ wbkt, const int* __restrict__ bbkt,
                     float* __restrict__ fbase) {
    const int bank = blockIdx.z;
    const int e    = blockIdx.y;
    const int n0   = blockIdx.x * 16;

    const float* W;
    const int *bkt, *cnt;
    float* dst;
    switch (bank) {
        case 0:  W = w_in;   bkt = wbkt; cnt = wcnt; dst = fbase + 0 * IVLEN; break;
        case 1:  W = w_out;  bkt = wbkt; cnt = wcnt; dst = fbase + 1 * IVLEN; break;
        case 2:  W = w_diag; bkt = wbkt; cnt = wcnt; dst = fbase + 2 * IVLEN; break;
        default: W = w_bias; bkt = bbkt; cnt = bcnt; dst = fbase + 3 * IVLEN; break;
    }
    W   += (size_t)e * DD * DD;
    bkt += e * NPAIR;
    const int count = cnt[e];

    const int lane = threadIdx.x;      // 0..31
    const int half = lane >> 4;        // 0: K=0/1, 1: K=2/3 (A operand halves)
    const int l15  = lane & 15;

    for (int mbase = 0; mbase < count; mbase += 16) {
        int mrem = count - mbase;
        if (mrem > 16) mrem = 16;

        // Lane l15 supplies A-row M=l15; clamp OOB rows to a valid bucket
        // entry (their D rows are never stored, see header comment).
        const int   mclamp  = (l15 < mrem) ? l15 : (mrem - 1);
        const int   rowPair = bkt[mbase + mclamp];
        const float* xrow   = x + (size_t)(rowPair >> 1) * DD;   // pair/K -> b

        v8f c = {};
        for (int k0 = 0; k0 < DD; k0 += 4) {
            if ((k0 & 63) == 0) {
                // Prefetch bank rows ~64 K-steps ahead (global_prefetch_b8).
                __builtin_prefetch(W + (size_t)((k0 + 64) & (DD - 1)) * DD + n0 + l15, 0, 1);
            }
            const int ka = k0 + 2 * half;
            const v2f a = *(const v2f*)(xrow + ka);      // 8B-aligned (ka even)
            v2f bm;
            bm.x = W[(size_t)ka * DD + n0 + l15];
            bm.y = W[(size_t)(ka + 1) * DD + n0 + l15];
            c = __builtin_amdgcn_wmma_f32_16x16x4_f32(
                    /*neg_a=*/false, a, /*neg_b=*/false, bm,
                    /*c_mod=*/(short)0, c, /*reuse_a=*/false, /*reuse_b=*/false);
        }

        // Scatter D tile: vgpr r holds row m = half*8 + r, column n0 + l15.
        // Lane m (0..15) already holds bucket[mbase+m] in rowPair: broadcast
        // it with a wave32 shuffle instead of reloading from memory.
        for (int r = 0; r < 8; ++r) {
            const int m    = half * 8 + r;
            const int pair = __shfl(rowPair, m, 32);
            if (m < mrem) {
                dst[(size_t)pair * DD + n0 + l15] = c[r];
            }
        }
    }
}

// ---------------------------------------------------------------------------
// Kernel C: rank-1 combine + diagonal + top-k mix + LayerNorm, one block
// (256 threads) per output row (b,i). Streams 256 MB of output.
// ---------------------------------------------------------------------------
__global__ __launch_bounds__(256)
void gen_combine_ln(const float* __restrict__ wp,
                    const float* __restrict__ iv,
                    const float* __restrict__ ov,
                    const float* __restrict__ dv,
                    float* __restrict__ out) {
    const int row = blockIdx.x;          // b*DD + i
    const int b   = row >> 9;
    const int i   = row & (DD - 1);
    const int t   = threadIdx.x;

    const int p0 = b * KK, p1 = b * KK + 1;
    const float w0 = wp[p0], w1 = wp[p1];
    const float c0 = w0 * iv[p0 * DD + i];
    const float c1 = w1 * iv[p1 * DD + i];
    const float dterm = w0 * dv[p0 * DD + i] + w1 * dv[p1 * DD + i];
    const float* o0 = ov + p0 * DD;
    const float* o1 = ov + p1 * DD;

    int j0 = t, j1 = t + 256;
    float v0 = c0 * o0[j0] + c1 * o1[j0];
    if (j0 == i) v0 += dterm;
    float v1 = c0 * o0[j1] + c1 * o1[j1];
    if (j1 == i) v1 += dterm;

    __shared__ float s1[256], s2[256];
    s1[t] = v0 + v1;
    s2[t] = v0 * v0 + v1 * v1;
    __syncthreads();
    for (int s = 128; s > 0; s >>= 1) {
        if (t < s) { s1[t] += s1[t + s]; s2[t] += s2[t + s]; }
        __syncthreads();
    }
    const float mu  = s1[0] * (1.0f / (float)DD);
    const float var = s2[0] * (1.0f / (float)DD) - mu * mu;
    const float rs  = rsqrtf(var + 1e-5f);

    float* orow = out + (size_t)row * DD;
    orow[j0] = (v0 - mu) * rs;
    orow[j1] = (v1 - mu) * rs;
}

// ---------------------------------------------------------------------------
// Kernel D: bias mix  bmix[b,o] = sum_k bp[b,k] * bv[b,k,o]
// ---------------------------------------------------------------------------
__global__ __launch_bounds__(256)
void gen_bias_mix(const float* __restrict__ bp,
                  const float* __restrict__ bv,
                  float* __restrict__ out2) {
    const int t = blockIdx.x * 256 + threadIdx.x;   // 0 .. B*D-1
    const int b = t >> 9;
    const int o = t & (DD - 1);
    out2[t] = bp[b * KK] * bv[(b * KK) * DD + o] +
              bp[b * KK + 1] * bv[(b * KK + 1) * DD + o];
}

extern "C" void kernel_launch(void* const* d_in, const int* in_sizes, int n_in,
                              void* d_out, int out_size, void* d_ws, size_t ws_size,
                              hipStream_t stream) {
    const float* x      = (const float*)d_in[0];
    const float* wprobs = (const float*)d_in[1];
    const float* bprobs = (const float*)d_in[2];
    const float* w_in   = (const float*)d_in[3];
    const float* w_out  = (const float*)d_in[4];
    const float* w_diag = (const float*)d_in[5];
    const float* w_bias = (const float*)d_in[6];
    const int*   widx   = (const int*)d_in[7];
    const int*   bidx   = (const int*)d_in[8];
    float* out = (float*)d_out;

    // Workspace layout: [wcnt 16][bcnt 16][wbkt 16*512][bbkt 16*512] ints,
    // then iv/ov/dv/bv (4 * 262144 floats). ~4.3 MB total.
    int* wcnt = (int*)d_ws;
    int* bcnt = wcnt + EE;
    int* wbkt = bcnt + EE;
    int* bbkt = wbkt + EE * NPAIR;
    float* fbase = (float*)(bbkt + EE * NPAIR);
    float* iv = fbase + 0 * IVLEN;
    float* ov = fbase + 1 * IVLEN;
    float* dv = fbase + 2 * IVLEN;
    float* bv = fbase + 3 * IVLEN;

    gen_bin<<<1, 32, 0, stream>>>(widx, bidx, wcnt, bcnt, wbkt, bbkt);

    dim3 gB(DD / 16, EE, 4);            // 32 N-tiles x 16 experts x 4 banks
    gen_expert_gemm<<<gB, 32, 0, stream>>>(x, w_in, w_out, w_diag, w_bias,
                                           wcnt, bcnt, wbkt, bbkt, fbase);

    gen_combine_ln<<<BB * DD, 256, 0, stream>>>(wprobs, iv, ov, dv, out);

    gen_bias_mix<<<(BB * DD) / 256, 256, 0, stream>>>(bprobs, bv,
                                                      out + (size_t)BB * DD * DD);
}